// Attention_41661182771360
// MI455X (gfx1250) — compile-verified
//
#include <hip/hip_runtime.h>

// ---------------------------------------------------------------------------
// Decode-step GQA attention for MI455X (gfx1250, wave32).
// Memory-bound: ~1.24 GB total traffic (1.07 GB KV cache) vs ~4.8 GFLOP
// => HBM roofline ~53 us at 23.3 TB/s. fp32 end-to-end; projections use
// V_WMMA_F32_16X16X4_F32. KV/weight streams are non-temporal (read once).
// ---------------------------------------------------------------------------

typedef __attribute__((ext_vector_type(2))) float v2f;
typedef __attribute__((ext_vector_type(4))) float v4f;
typedef __attribute__((ext_vector_type(8))) float v8f;

#define BATCH 32
#define DIMK  4096      // K dimension of every GEMM here (DIM and NH*HD)
#define NH    32
#define NKV   8
#define NREP  4
#define HD    128
#define TLEN  4096      // start_pos(4095) + 1
#define KSPLIT 4        // waves per GEMM block, splitting the K range

__device__ __forceinline__ v8f wmma_f32_4(v2f a, v2f b, v8f c) {
  // 8-arg pattern: (neg_a, A, neg_b, B, c_mod, C, reuse_a, reuse_b)
  return __builtin_amdgcn_wmma_f32_16x16x4_f32(false, a, false, b, (short)0, c,
                                               false, false);
}

// ---------------------------------------------------------------------------
// C[32][N] = A[32][4096] * W[4096][N]   (fp32, row-major)
// 32-column slab per 128-thread block => every 128B weight line is consumed
// by exactly one workgroup. 4 waves split K, LDS tree-combine at the end.
// A-matrix VGPR layout (ISA 7.12.2, 32-bit A 16x4):
//   lane l in [0,16):  VGPR0=A[l][k+0],    VGPR1=A[l][k+1]
//   lane l in [16,32): VGPR0=A[l-16][k+2], VGPR1=A[l-16][k+3]
// B (4x16): VGPR0 = B[k+2*hi][n0+l], VGPR1 = B[k+1+2*hi][n0+l]
// D: c[r] = D[r + 8*hi][n0+l]
// ---------------------------------------------------------------------------
__global__ void gemm_m32_wmma(const float* __restrict__ A,
                              const float* __restrict__ W,
                              float* __restrict__ Dst, int N) {
  const int lane = threadIdx.x & 31;
  const int wave = threadIdx.x >> 5;      // 0..3, K-split
  const int l    = lane & 15;
  const int hi   = lane >> 4;
  const int n0   = blockIdx.x * 32;       // 32-column slab (one full line)

  const float* a0p = A + (size_t)l * DIMK;          // rows 0..15
  const float* a1p = A + (size_t)(16 + l) * DIMK;   // rows 16..31
  const float* wp0 = W + (size_t)n0 + l;            // columns n0..n0+15
  const float* wp1 = wp0 + 16;                      // columns n0+16..n0+31

  const int kbeg = wave * (DIMK / KSPLIT);
  const int kend = kbeg + (DIMK / KSPLIT);

  v8f c00 = {};   // rows 0..15  x cols n0..n0+15
  v8f c10 = {};   // rows 16..31 x cols n0..n0+15
  v8f c01 = {};   // rows 0..15  x cols n0+16..n0+31
  v8f c11 = {};   // rows 16..31 x cols n0+16..n0+31
#pragma unroll 4
  for (int k = kbeg; k < kend; k += 4) {
    const int ka = k + 2 * hi;
    const v2f a0 = *(const v2f*)(a0p + ka);
    const v2f a1 = *(const v2f*)(a1p + ka);
    const size_t r0 = (size_t)ka * N;
    const size_t r1 = (size_t)(ka + 1) * N;
    v2f b0, b1;                       // weights streamed once -> NT hint
    b0.x = __builtin_nontemporal_load(wp0 + r0);
    b0.y = __builtin_nontemporal_load(wp0 + r1);
    b1.x = __builtin_nontemporal_load(wp1 + r0);
    b1.y = __builtin_nontemporal_load(wp1 + r1);
    c00 = wmma_f32_4(a0, b0, c00);
    c10 = wmma_f32_4(a1, b0, c10);
    c01 = wmma_f32_4(a0, b1, c01);
    c11 = wmma_f32_4(a1, b1, c11);
  }

  __shared__ float red[KSPLIT - 1][4][32][8];   // [wave-1][tile][lane][r]
  if (wave > 0) {
#pragma unroll
    for (int r = 0; r < 8; ++r) {
      red[wave - 1][0][lane][r] = c00[r];
      red[wave - 1][1][lane][r] = c10[r];
      red[wave - 1][2][lane][r] = c01[r];
      red[wave - 1][3][lane][r] = c11[r];
    }
  }
  __syncthreads();
  if (wave == 0) {
#pragma unroll
    for (int r = 0; r < 8; ++r) {
      float s00 = c00[r], s10 = c10[r], s01 = c01[r], s11 = c11[r];
#pragma unroll
      for (int w = 0; w < KSPLIT - 1; ++w) {
        s00 += red[w][0][lane][r];
        s10 += red[w][1][lane][r];
        s01 += red[w][2][lane][r];
        s11 += red[w][3][lane][r];
      }
      const int m = r + 8 * hi;
      Dst[(size_t)m * N + n0 + l]             = s00;
      Dst[(size_t)(16 + m) * N + n0 + l]      = s10;
      Dst[(size_t)m * N + n0 + 16 + l]        = s01;
      Dst[(size_t)(16 + m) * N + n0 + 16 + l] = s11;
    }
  }
}

// ---------------------------------------------------------------------------
// RoPE at position 4095 applied in-place to q (B*NH heads) and k (B*NKV heads)
// ---------------------------------------------------------------------------
__global__ void rope_kernel(float* __restrict__ qbuf, float* __restrict__ kbuf,
                            const float* __restrict__ fc,
                            const float* __restrict__ fs) {
  const int tid  = blockIdx.x * blockDim.x + threadIdx.x;
  const int p    = tid & 63;              // pair index within head (HD/2)
  const int inst = tid >> 6;              // which head instance
  if (inst >= BATCH * (NH + NKV)) return;
  float* base = (inst < BATCH * NH)
                    ? qbuf + (size_t)inst * HD
                    : kbuf + (size_t)(inst - BATCH * NH) * HD;
  const float x0 = base[2 * p], x1 = base[2 * p + 1];
  const float c = fc[p], s = fs[p];
  base[2 * p]     = x0 * c - x1 * s;
  base[2 * p + 1] = x0 * s + x1 * c;
}

// ---------------------------------------------------------------------------
// One workgroup (256 threads) per (batch b, kv-group g):
//   scores[4][4096] -> softmax -> out[4][128]
// Position 4095 K/V comes from workspace (inputs are never mutated).
// KV cache is streamed once -> non-temporal loads + row prefetch.
// ---------------------------------------------------------------------------
__global__ void attn_kernel(const float* __restrict__ cache_k,
                            const float* __restrict__ cache_v,
                            const float* __restrict__ qbuf,
                            const float* __restrict__ kbuf,
                            const float* __restrict__ vbuf,
                            float* __restrict__ attn_out) {
  const int b = blockIdx.x >> 3;          // / NKV
  const int g = blockIdx.x & 7;           // % NKV
  const int tid = threadIdx.x;            // 0..255

  __shared__ __align__(16) float s_scores[NREP][TLEN];   // 64 KB
  __shared__ __align__(16) float s_q[NREP][HD];          // 2 KB
  __shared__ float s_red[NREP][256];                     // 4 KB
  __shared__ float s_fin[NREP];

  // stage the 4 query heads of this group
  const float* qsrc = qbuf + ((size_t)b * NH + g * NREP) * HD;
  for (int i = tid; i < NREP * HD; i += 256) s_q[i / HD][i % HD] = qsrc[i];
  __syncthreads();

  const float scale = 0.08838834764831845f;  // 1/sqrt(128)
  const float* knew = kbuf + ((size_t)b * NKV + g) * HD;
  const float* vnew = vbuf + ((size_t)b * NKV + g) * HD;
  const size_t krow0 = ((size_t)b * TLEN * NKV + g) * HD;  // + t*NKV*HD

  // ---- phase 1: scores = q . K, track row maxima --------------------------
  float lmax[NREP];
#pragma unroll
  for (int r = 0; r < NREP; ++r) lmax[r] = -3.0e38f;

  for (int t = tid; t < TLEN; t += 256) {
    const float* kp = (t < TLEN - 1) ? cache_k + krow0 + (size_t)t * NKV * HD
                                     : knew;
    if (t + 256 < TLEN - 1)   // cover the 1MB jump to this thread's next row
      __builtin_prefetch(cache_k + krow0 + (size_t)(t + 256) * NKV * HD, 0, 0);
    float acc[NREP] = {0.f, 0.f, 0.f, 0.f};
#pragma unroll 8
    for (int d = 0; d < HD; d += 4) {
      const v4f kv = __builtin_nontemporal_load((const v4f*)(kp + d));
#pragma unroll
      for (int r = 0; r < NREP; ++r) {
        const v4f qv = *(const v4f*)(&s_q[r][d]);
        acc[r] += qv.x * kv.x + qv.y * kv.y + qv.z * kv.z + qv.w * kv.w;
      }
    }
#pragma unroll
    for (int r = 0; r < NREP; ++r) {
      const float sc = acc[r] * scale;
      s_scores[r][t] = sc;
      lmax[r] = fmaxf(lmax[r], sc);
    }
  }

  // row-max reduction over 256 threads
#pragma unroll
  for (int r = 0; r < NREP; ++r) s_red[r][tid] = lmax[r];
  __syncthreads();
  for (int st = 128; st > 0; st >>= 1) {
    if (tid < st)
#pragma unroll
      for (int r = 0; r < NREP; ++r)
        s_red[r][tid] = fmaxf(s_red[r][tid], s_red[r][tid + st]);
    __syncthreads();
  }
  if (tid < NREP) s_fin[tid] = s_red[tid][0];
  __syncthreads();
  float rmax[NREP];
#pragma unroll
  for (int r = 0; r < NREP; ++r) rmax[r] = s_fin[r];
  __syncthreads();

  // ---- phase 2: exponentiate in LDS, accumulate row sums ------------------
  float lsum[NREP] = {0.f, 0.f, 0.f, 0.f};
  for (int t = tid; t < TLEN; t += 256) {
#pragma unroll
    for (int r = 0; r < NREP; ++r) {
      const float e = __expf(s_scores[r][t] - rmax[r]);
      s_scores[r][t] = e;
      lsum[r] += e;
    }
  }
#pragma unroll
  for (int r = 0; r < NREP; ++r) s_red[r][tid] = lsum[r];
  __syncthreads();
  for (int st = 128; st > 0; st >>= 1) {
    if (tid < st)
#pragma unroll
      for (int r = 0; r < NREP; ++r) s_red[r][tid] += s_red[r][tid + st];
    __syncthreads();
  }
  if (tid < NREP) s_fin[tid] = 1.0f / s_red[tid][0];
  __syncthreads();

  // ---- phase 3: out = p @ V, each thread owns column d, halves split t ----
  const int d    = tid & (HD - 1);
  const int half = tid >> 7;
  float acc[NREP] = {0.f, 0.f, 0.f, 0.f};
  const int tbeg = half * (TLEN / 2);
  const int tend = tbeg + (TLEN / 2);
  const size_t vrow0 = ((size_t)b * TLEN * NKV + g) * HD;
  for (int t = tbeg; t < tend; ++t) {
    const float* vp = (t < TLEN - 1) ? cache_v + vrow0 + (size_t)t * NKV * HD
                                     : vnew;
    const float v = __builtin_nontemporal_load(vp + d);
#pragma unroll
    for (int r = 0; r < NREP; ++r) acc[r] += s_scores[r][t] * v;
  }
#pragma unroll
  for (int r = 0; r < NREP; ++r) s_red[r][tid] = acc[r];
  __syncthreads();
  if (half == 0) {
#pragma unroll
    for (int r = 0; r < NREP; ++r) {
      const float o = (s_red[r][d] + s_red[r][d + 128]) * s_fin[r];
      attn_out[((size_t)b * NH + g * NREP + r) * HD + d] = o;
    }
  }
}

// ---------------------------------------------------------------------------
extern "C" void kernel_launch(void* const* d_in, const int* in_sizes, int n_in,
                              void* d_out, int out_size, void* d_ws,
                              size_t ws_size, hipStream_t stream) {
  const float* x  = (const float*)d_in[0];
  const float* ck = (const float*)d_in[1];
  const float* cv = (const float*)d_in[2];
  const float* wq = (const float*)d_in[3];
  const float* wk = (const float*)d_in[4];
  const float* wv = (const float*)d_in[5];
  const float* wo = (const float*)d_in[6];
  const float* fc = (const float*)d_in[7];
  const float* fs = (const float*)d_in[8];
  float* out = (float*)d_out;

  // workspace: q[32][4096] | k_new[32][1024] | v_new[32][1024] | attn[32][4096]
  float* qbuf = (float*)d_ws;
  float* kbuf = qbuf + (size_t)BATCH * NH * HD;
  float* vbuf = kbuf + (size_t)BATCH * NKV * HD;
  float* abuf = vbuf + (size_t)BATCH * NKV * HD;

  // QKV projections (fp32 WMMA), 32-column slabs, 4-way K-split
  gemm_m32_wmma<<<(NH * HD) / 32, 32 * KSPLIT, 0, stream>>>(x, wq, qbuf, NH * HD);
  gemm_m32_wmma<<<(NKV * HD) / 32, 32 * KSPLIT, 0, stream>>>(x, wk, kbuf, NKV * HD);
  gemm_m32_wmma<<<(NKV * HD) / 32, 32 * KSPLIT, 0, stream>>>(x, wv, vbuf, NKV * HD);

  // RoPE on q and k (position 4095)
  const int rope_threads = BATCH * (NH + NKV) * (HD / 2);
  rope_kernel<<<(rope_threads + 255) / 256, 256, 0, stream>>>(qbuf, kbuf, fc, fs);

  // attention: one WG per (b, kv-head)
  attn_kernel<<<BATCH * NKV, 256, 0, stream>>>(ck, cv, qbuf, kbuf, vbuf, abuf);

  // output projection into d_out
  gemm_m32_wmma<<<DIMK / 32, 32 * KSPLIT, 0, stream>>>(abuf, wo, out, DIMK);
}